// GHConv_27238682591748
// MI455X (gfx1250) — compile-verified
//
#include <hip/hip_runtime.h>
#include <math.h>

typedef __attribute__((ext_vector_type(2))) float v2f;
typedef __attribute__((ext_vector_type(8))) float v8f;

namespace {
constexpr int B_ = 8;
constexpr int N_ = 10000;
constexpr int E_ = 320000;
constexpr int H_ = 128;
constexpr int ROWS = B_ * N_;     // 80000 rows of flattened [B*N, H]
constexpr int NEDGE = B_ * E_;    // 2,560,000 edges
constexpr int SCAN_T = 640;       // scan threads
constexpr int SCAN_PER = ROWS / SCAN_T;  // 125 rows/thread (exact)
}

// ---------------------------------------------------------------------------
// Zero 4-byte words (count + deg regions only; everything else fully written).
// ---------------------------------------------------------------------------
__global__ void zero_b32(unsigned int* __restrict__ p, int n) {
  int i = blockIdx.x * blockDim.x + threadIdx.x;
  const int stride = gridDim.x * blockDim.x;
  for (; i < n; i += stride) p[i] = 0u;
}

// ---------------------------------------------------------------------------
// Per-row edge count (for CSR) + degree = segment_sum(adj_val over src).
// ---------------------------------------------------------------------------
__global__ void count_deg_kernel(const float* __restrict__ adj_val,
                                 const int* __restrict__ src,
                                 int* __restrict__ count,
                                 float* __restrict__ deg) {
  const int t = blockIdx.x * blockDim.x + threadIdx.x;
  if (t >= NEDGE) return;
  const int b = t / E_;
  const int r = b * N_ + src[t];
  atomicAdd(&count[r], 1);
  atomicAdd(&deg[r], adj_val[t]);
}

// ---------------------------------------------------------------------------
// Single-block exclusive scan of count[ROWS] -> rowStart[ROWS+1] and cursor.
// 640 threads x 125 rows each; Hillis-Steele scan of 640 partials in LDS.
// ---------------------------------------------------------------------------
__global__ void scan_kernel(const int* __restrict__ count,
                            int* __restrict__ rowStart,
                            int* __restrict__ cursor) {
  __shared__ int part[SCAN_T];
  const int t = threadIdx.x;
  const int base = t * SCAN_PER;
  int s = 0;
#pragma unroll 5
  for (int i = 0; i < SCAN_PER; ++i) s += count[base + i];
  part[t] = s;
  __syncthreads();
  for (int off = 1; off < SCAN_T; off <<= 1) {
    const int v = (t >= off) ? part[t - off] : 0;
    __syncthreads();
    part[t] += v;
    __syncthreads();
  }
  int run = part[t] - s;  // exclusive prefix for this thread's chunk
#pragma unroll 5
  for (int i = 0; i < SCAN_PER; ++i) {
    rowStart[base + i] = run;
    cursor[base + i] = run;
    run += count[base + i];
  }
  if (t == SCAN_T - 1) rowStart[ROWS] = run;  // == NEDGE
}

// ---------------------------------------------------------------------------
// CSR fill: place each edge at its row's next slot; payload = (dst_row, val).
// ---------------------------------------------------------------------------
__global__ void fill_kernel(const float* __restrict__ adj_val,
                            const int* __restrict__ src,
                            const int* __restrict__ dst,
                            int* __restrict__ cursor,
                            int2* __restrict__ edges) {
  const int t = blockIdx.x * blockDim.x + threadIdx.x;
  if (t >= NEDGE) return;
  const int b = t / E_;
  const int r = b * N_ + src[t];
  const int pos = atomicAdd(&cursor[r], 1);
  edges[pos] = make_int2(b * N_ + dst[t], __float_as_int(adj_val[t]));
}

// ---------------------------------------------------------------------------
// Fused triple GEMM via V_WMMA_F32_16X16X4_F32 (fp32 matrix core path).
// One wave owns a 16-row tile of x; A fragments loaded once (32 x float2 per
// lane) and reused across 8 N-tiles x 3 weights. Epilogue fuses:
//   fhom = (x@theta)*norm ; gate = sigmoid(x@W_t + b_t) ;
//   d_out = (1-gate)*(x@W_h)   (staged for the gather pass).
// Fragment layouts per ISA 7.12.2 (32-bit, wave32):
//   A (16x4):  lane l -> row m = l%16, cols k = 2*(l/16)+{0,1} (contig float2)
//   B (4x16):  lane l -> rows k = 2*(l/16)+{0,1}, col n = l%16
//   C (16x16): c[i]   -> row  i + 8*(l/16), col n = l%16
// ---------------------------------------------------------------------------
__global__ __launch_bounds__(256) void gemm3_wmma_kernel(
    const float* __restrict__ x, const float* __restrict__ theta,
    const float* __restrict__ Wh, const float* __restrict__ Wt,
    const float* __restrict__ bt, const float* __restrict__ deg,
    float* __restrict__ fhom, float* __restrict__ gate,
    float* __restrict__ outp) {
  const int lane = threadIdx.x & 31;
  const int wave = threadIdx.x >> 5;
  const int tile = blockIdx.x * (blockDim.x >> 5) + wave;
  const int row0 = tile * 16;
  if (row0 >= ROWS) return;  // wave-uniform; never taken with exact grid

  const int m  = lane & 15;
  const int kb = (lane >> 4) << 1;  // 0 or 2

  v2f a[32];
  const float* xrow = x + (size_t)(row0 + m) * H_;
#pragma unroll
  for (int kk = 0; kk < 32; ++kk)
    a[kk] = *(const v2f*)(xrow + kk * 4 + kb);

  const int roff = (lane >> 4) * 8;

#pragma unroll 1
  for (int nt = 0; nt < 8; ++nt) {
    const int n = nt * 16 + m;
    v8f ct = {};  // x @ theta
    v8f ch = {};  // x @ W_h
    v8f cg = {};  // x @ W_t
#pragma unroll
    for (int kk = 0; kk < 32; ++kk) {
      const int k = kk * 4 + kb;
      v2f bth = {theta[k * H_ + n], theta[(k + 1) * H_ + n]};
      v2f bwh = {Wh[k * H_ + n],    Wh[(k + 1) * H_ + n]};
      v2f bwt = {Wt[k * H_ + n],    Wt[(k + 1) * H_ + n]};
      ct = __builtin_amdgcn_wmma_f32_16x16x4_f32(false, a[kk], false, bth,
                                                 (short)0, ct, false, false);
      ch = __builtin_amdgcn_wmma_f32_16x16x4_f32(false, a[kk], false, bwh,
                                                 (short)0, ch, false, false);
      cg = __builtin_amdgcn_wmma_f32_16x16x4_f32(false, a[kk], false, bwt,
                                                 (short)0, cg, false, false);
    }
    const float bias = bt[n];
#pragma unroll
    for (int i = 0; i < 8; ++i) {
      const int r = row0 + roff + i;
      const float nr = rsqrtf(deg[r] + 1e-6f);
      const float g  = 1.0f / (1.0f + __expf(-(cg[i] + bias)));
      const size_t o = (size_t)r * H_ + n;
      fhom[o] = ct[i] * nr;          // pre-normalized f_hom (dst-side norm)
      gate[o] = g;
      outp[o] = (1.0f - g) * ch[i];  // (1-gate)*f_het staged into d_out
    }
  }
}

// ---------------------------------------------------------------------------
// SpMM as CSR gather + fused combine. One wave per output row r:
//   acc = sum over edges(r) of val * fhom[dst_row]   (float4 per lane)
//   out[r] = gate[r] * (acc * norm[r]) + staged (1-gate)*f_het  (RMW of d_out)
// Zero atomics; fhom gathers are 512B rows resident in the 192MB L2.
// ---------------------------------------------------------------------------
__global__ __launch_bounds__(256) void spmm_gather_combine_kernel(
    const float* __restrict__ fhom, const int2* __restrict__ edges,
    const int* __restrict__ rowStart, const float* __restrict__ deg,
    const float* __restrict__ gate, float* __restrict__ outp) {
  const int lane = threadIdx.x & 31;
  const int wrow = blockIdx.x * (blockDim.x >> 5) + (threadIdx.x >> 5);
  const int r = __builtin_amdgcn_readfirstlane(wrow);  // force wave-uniform
  if (r >= ROWS) return;
  const int beg = rowStart[r];
  const int end = rowStart[r + 1];
  float4 acc = make_float4(0.f, 0.f, 0.f, 0.f);
  for (int idx = beg; idx < end; ++idx) {
    const int2 e = edges[idx];  // wave-uniform (broadcast by TA / L1)
    const float v = __int_as_float(e.y);
    const float4 y = ((const float4*)(fhom + (size_t)e.x * H_))[lane];
    acc.x = fmaf(v, y.x, acc.x);
    acc.y = fmaf(v, y.y, acc.y);
    acc.z = fmaf(v, y.z, acc.z);
    acc.w = fmaf(v, y.w, acc.w);
  }
  const float nr = rsqrtf(deg[r] + 1e-6f);
  const size_t o4 = (size_t)r * (H_ / 4) + lane;
  const float4 g = ((const float4*)gate)[o4];
  float4 st = ((float4*)outp)[o4];  // holds (1-gate)*f_het
  st.x = fmaf(g.x * nr, acc.x, st.x);
  st.y = fmaf(g.y * nr, acc.y, st.y);
  st.z = fmaf(g.z * nr, acc.z, st.z);
  st.w = fmaf(g.w * nr, acc.w, st.w);
  ((float4*)outp)[o4] = st;
}

// ---------------------------------------------------------------------------
extern "C" void kernel_launch(void* const* d_in, const int* in_sizes, int n_in,
                              void* d_out, int out_size, void* d_ws,
                              size_t ws_size, hipStream_t stream) {
  const float* x       = (const float*)d_in[0];
  const float* adj_val = (const float*)d_in[1];
  const int*   src     = (const int*)d_in[2];
  const int*   dst     = (const int*)d_in[3];
  const float* W_t     = (const float*)d_in[4];
  const float* b_t     = (const float*)d_in[5];
  const float* W_h     = (const float*)d_in[6];
  const float* theta   = (const float*)d_in[7];
  float* out = (float*)d_out;

  // Workspace layout in 4-byte words (base is >=256B aligned):
  //   count[ROWS] | deg[ROWS] | rowStart[ROWS+1] | pad | cursor[ROWS] |
  //   edges[NEDGE int2] | fhom[ROWS*H] (16B aligned) | gate[ROWS*H]
  unsigned int* w32 = (unsigned int*)d_ws;
  int*   count    = (int*)w32;                        // off 0
  float* deg      = (float*)(w32 + ROWS);             // off ROWS
  int*   rowStart = (int*)(w32 + 2 * ROWS);           // off 2*ROWS, ROWS+1 ints
  int*   cursor   = (int*)(w32 + 3 * ROWS + 2);       // even offset
  int2*  edges    = (int2*)(w32 + 4 * ROWS + 2);      // 8B aligned
  size_t off = (size_t)4 * ROWS + 2 + 2 * (size_t)NEDGE;
  off = (off + 3) & ~(size_t)3;                       // 16B align for float4
  float* fhom = (float*)(w32 + off);
  float* gate = fhom + (size_t)ROWS * H_;

  // 1) clear count + deg (contiguous 2*ROWS words)
  zero_b32<<<(2 * ROWS + 255) / 256, 256, 0, stream>>>(w32, 2 * ROWS);
  // 2) per-row counts + degrees
  count_deg_kernel<<<NEDGE / 256, 256, 0, stream>>>(adj_val, src, count, deg);
  // 3) exclusive scan -> rowStart, cursor
  scan_kernel<<<1, SCAN_T, 0, stream>>>(count, rowStart, cursor);
  // 4) CSR fill
  fill_kernel<<<NEDGE / 256, 256, 0, stream>>>(adj_val, src, dst, cursor,
                                               edges);
  // 5) fused triple GEMM (5000 tiles = 625 blocks x 8 waves, EXEC all-ones)
  gemm3_wmma_kernel<<<(ROWS / 16) / 8, 256, 0, stream>>>(
      x, theta, W_h, W_t, b_t, deg, fhom, gate, out);
  // 6) CSR gather + final combine (80000 rows = 10000 blocks x 8 waves)
  spmm_gather_combine_kernel<<<ROWS / 8, 256, 0, stream>>>(
      fhom, edges, rowStart, deg, gate, out);
}